// AngularPenaltySMLoss_42150809043622
// MI455X (gfx1250) — compile-verified
//
#include <hip/hip_runtime.h>
#include <cstdint>
#include <cstddef>

// ---------------------------------------------------------------------------
// ArcFace loss, MI455X (gfx1250).
// Bandwidth-bound streaming reduction: 327.7 MB @ 23.3 TB/s ~= 14 us floor.
// Data path: CDNA5 async global->LDS copy engine, double-buffered per row,
// non-temporal (stream-once, > L2 capacity, zero reuse).
// ---------------------------------------------------------------------------

constexpr int BLOCK  = 256;    // 8 x wave32
constexpr int CHUNK4 = 1024;   // float4 slots per chunk buffer (4096 floats, 16 KB)
constexpr int CHUNKF = 4 * CHUNK4;

// Async copy of 16 bytes per lane from global memory at sbase + goff to the LDS address in lds_addr.
// Tracked by ASYNCcnt; retires in order with other async loads of this wave.
// TH_LOAD_NT: stream-once data, do not thrash caches.
__device__ __forceinline__ void async_copy_b128_nt(unsigned lds_addr, unsigned goff,
                                                   const float* sbase) {
  asm volatile("global_load_async_to_lds_b128 %0, %1, %2 th:TH_LOAD_NT"
               :
               : "v"(lds_addr), "v"(goff), "s"(sbase)
               : "memory");
}

__device__ __forceinline__ void wait_async_0() {
  asm volatile("s_wait_asynccnt 0x0" ::: "memory");
}
__device__ __forceinline__ void wait_async_4() {
  asm volatile("s_wait_asynccnt 0x4" ::: "memory");
}
__device__ __forceinline__ void wait_ds_0() {
  asm volatile("s_wait_dscnt 0x0" ::: "memory");
}

__global__ void __launch_bounds__(BLOCK)
arcface_row_kernel(const float* __restrict__ wf,
                   const int*   __restrict__ labels,
                   float*       __restrict__ rowL,
                   int B, int C) {
  __shared__ float4 buf[2][CHUNK4];          // 32 KB double buffer
  __shared__ float  red[BLOCK / 32];

  const int    row  = blockIdx.x;
  const int    tid  = threadIdx.x;
  const float* rowp = wf + (size_t)row * (size_t)C;

  const int      nchunks = (C + CHUNKF - 1) / CHUNKF;
  const unsigned maxOff  = (unsigned)(C * 4 - 16);   // last aligned float4 of row

  auto issue = [&](int c, int b) {
    // Formal safety: prior ds_loads from this buffer must drain before the
    // async engine may overwrite it.
    wait_ds_0();
#pragma unroll
    for (int k = 0; k < 4; ++k) {
      const int      slot  = tid + k * BLOCK;                   // 0..1023
      unsigned       goff  = (unsigned)(c * CHUNKF + slot * 4) * 4u;
      if (goff > maxOff) goff = maxOff;                         // clamp, never OOB
      const unsigned laddr = (unsigned)(size_t)(&buf[b][slot]); // flat low32 == LDS offset
      async_copy_b128_nt(laddr, goff, rowp);
    }
  };

  float acc = 0.0f;
  issue(0, 0);
  for (int c = 0; c < nchunks; ++c) {
    if (c + 1 < nchunks) {
      issue(c + 1, (c + 1) & 1);   // overlap next chunk's DMA with this compute
      wait_async_4();              // async loads retire in order: oldest 4 landed
    } else {
      wait_async_0();
    }
    if (c * CHUNKF + CHUNKF <= C) {
      // Full chunk: no per-slot bounds check (block-uniform branch).
#pragma unroll
      for (int k = 0; k < 4; ++k) {
        const int    slot = tid + k * BLOCK;
        const float4 v    = buf[c & 1][slot];      // ds_load_b128 from own slot
        acc += __expf(64.0f * v.x) + __expf(64.0f * v.y) +
               __expf(64.0f * v.z) + __expf(64.0f * v.w);
      }
    } else {
      // Tail chunk: C*4 is 16-aligned, so validity is all-or-nothing per float4.
#pragma unroll
      for (int k = 0; k < 4; ++k) {
        const int slot = tid + k * BLOCK;
        const int col  = c * CHUNKF + slot * 4;
        if (col < C) {
          const float4 v = buf[c & 1][slot];
          acc += __expf(64.0f * v.x) + __expf(64.0f * v.y) +
                 __expf(64.0f * v.z) + __expf(64.0f * v.w);
        }
      }
    }
  }

  // Block reduction of the exp-sum.
  for (int o = 16; o > 0; o >>= 1) acc += __shfl_down(acc, o, 32);
  const int lane = tid & 31, wv = tid >> 5;
  if (lane == 0) red[wv] = acc;
  __syncthreads();

  if (tid == 0) {
    float rs = 0.0f;
#pragma unroll
    for (int w = 0; w < BLOCK / 32; ++w) rs += red[w];

    const int   lab = labels[row];
    const float tgt = rowp[lab];
    // numerator = S*cos(acos(clip(tgt)) + M) = S*(c*cosM - sqrt(1-c^2)*sinM)
    float c0 = fminf(fmaxf(tgt, -1.0f + 1e-7f), 1.0f - 1e-7f);
    const float cosM = 0.8775825618903728f;   // cos(0.5)
    const float sinM = 0.4794255386042030f;   // sin(0.5)
    const float num  = 64.0f * (c0 * cosM - sqrtf(fmaxf(1.0f - c0 * c0, 0.0f)) * sinM);
    const float excl = rs - __expf(64.0f * tgt);       // unclamped target, as reference
    const float den  = __expf(num) + excl;
    rowL[row] = logf(den) - num;                        // = -L_i
  }
}

__global__ void __launch_bounds__(1024)
reduce_mean_kernel(const float* __restrict__ rowL, float* __restrict__ out, int B) {
  __shared__ float red[32];
  float acc = 0.0f;
  for (int i = threadIdx.x; i < B; i += blockDim.x) acc += rowL[i];
  for (int o = 16; o > 0; o >>= 1) acc += __shfl_down(acc, o, 32);
  const int lane = threadIdx.x & 31, wv = threadIdx.x >> 5;
  if (lane == 0) red[wv] = acc;
  __syncthreads();
  if (wv == 0) {
    const int nw = blockDim.x >> 5;
    float v = (lane < nw) ? red[lane] : 0.0f;
    for (int o = 16; o > 0; o >>= 1) v += __shfl_down(v, o, 32);
    if (lane == 0) out[0] = v / (float)B;
  }
}

extern "C" void kernel_launch(void* const* d_in, const int* in_sizes, int n_in,
                              void* d_out, int out_size, void* d_ws, size_t ws_size,
                              hipStream_t stream) {
  const float* wf     = (const float*)d_in[0];
  const int*   labels = (const int*)d_in[1];    // jax downgrades int64->int32 (x64 off)
  const int B = in_sizes[1];                    // 8192
  const int C = in_sizes[0] / B;                // 10000

  float* rowL = (float*)d_ws;                   // B floats of scratch (32 KB)

  arcface_row_kernel<<<B, BLOCK, 0, stream>>>(wf, labels, rowL, B, C);
  reduce_mean_kernel<<<1, 1024, 0, stream>>>(rowL, (float*)d_out, B);
}